// MultiHeadGraphAttention_32091995636256
// MI455X (gfx1250) — compile-verified
//
#include <hip/hip_runtime.h>
#include <math.h>

#define N_NODES 50000
#define N_EDGES 800000
#define DMODEL  128
#define NHEADS  8

typedef float v2f __attribute__((ext_vector_type(2)));
typedef float v8f __attribute__((ext_vector_type(8)));

// ---------------------------------------------------------------------------
// WMMA GEMM: Y[M,128] = X[M,128] @ W[128,128] + bias   (M % 16 == 0)
// One 256-thread block = 8 waves; block handles one 16-row tile, each wave
// produces one 16x16 output tile using V_WMMA_F32_16X16X4_F32 (K stepped by 4).
// ---------------------------------------------------------------------------
__global__ __launch_bounds__(256) void gemm128_wmma(const float* __restrict__ X,
                                                    const float* __restrict__ W,
                                                    const float* __restrict__ bias,
                                                    float* __restrict__ Y, int M)
{
    __shared__ float As[16][132];   // padded to avoid 16-way bank conflicts
    const int tid = threadIdx.x;
    const int m0  = blockIdx.x * 16;

    // cooperative load of the 16x128 A tile (coalesced)
    #pragma unroll
    for (int i = 0; i < 8; ++i) {
        int lin = i * 256 + tid;
        int r = lin >> 7, c = lin & 127;
        As[r][c] = X[(size_t)(m0 + r) * DMODEL + c];
    }
    __syncthreads();

    const int wave = tid >> 5;
    const int lane = tid & 31;
    const int half = lane >> 4;      // 0: K pair {0,1}, 1: K pair {2,3}
    const int mr   = lane & 15;      // M (for A) / N (for B,C) within tile
    const int n0   = wave * 16;

    const float bv = bias[n0 + mr];  // bias depends on column only
    v8f acc = {bv, bv, bv, bv, bv, bv, bv, bv};

    #pragma unroll 4
    for (int k = 0; k < DMODEL; k += 4) {
        v2f a, b;
        a.x = As[mr][k + 2 * half + 0];
        a.y = As[mr][k + 2 * half + 1];
        b.x = W[(size_t)(k + 2 * half + 0) * DMODEL + n0 + mr];
        b.y = W[(size_t)(k + 2 * half + 1) * DMODEL + n0 + mr];
        acc = __builtin_amdgcn_wmma_f32_16x16x4_f32(false, a, false, b,
                                                    (short)0, acc, false, false);
    }

    #pragma unroll
    for (int r = 0; r < 8; ++r) {
        int row = m0 + r + half * 8;
        Y[(size_t)row * DMODEL + n0 + mr] = acc[r];
    }
}

// ---------------------------------------------------------------------------
// Node-level edge-bias projection: out[n,h] = x[n,:] @ WePart[:,h] (+ be[h])
// WePart is a [128,8] row-major slice of We.  One wave per node.
// ---------------------------------------------------------------------------
__global__ __launch_bounds__(256) void edge_bias_proj(const float* __restrict__ X,
                                                      const float* __restrict__ WePart,
                                                      const float* __restrict__ be,
                                                      float* __restrict__ out, int n)
{
    int node = blockIdx.x * (blockDim.x >> 5) + (threadIdx.x >> 5);
    int lane = threadIdx.x & 31;
    if (node >= n) return;

    const float4 xv = ((const float4*)(X + (size_t)node * DMODEL))[lane];
    float res = 0.f;
    #pragma unroll
    for (int h = 0; h < NHEADS; ++h) {
        float p = xv.x * WePart[(4 * lane + 0) * NHEADS + h]
                + xv.y * WePart[(4 * lane + 1) * NHEADS + h]
                + xv.z * WePart[(4 * lane + 2) * NHEADS + h]
                + xv.w * WePart[(4 * lane + 3) * NHEADS + h];
        #pragma unroll
        for (int off = 16; off > 0; off >>= 1) p += __shfl_xor(p, off, 32);
        if (lane == h) res = p;
    }
    if (lane < NHEADS)
        out[(size_t)node * NHEADS + lane] = res + (be ? be[lane] : 0.f);
}

// ---------------------------------------------------------------------------
__global__ void fill_f32(float* __restrict__ p, float val, int n)
{
    int i = blockIdx.x * blockDim.x + threadIdx.x;
    if (i < n) p[i] = val;
}

// ---------------------------------------------------------------------------
// Pass A: per-edge multi-head scores + segment max.  One wave per edge.
// lane covers d = 4*lane..4*lane+3 ; head = lane>>2 ; reduce over 4 lanes.
// ---------------------------------------------------------------------------
__global__ __launch_bounds__(256) void edge_scores_max(const int* __restrict__ ei,
    const float* __restrict__ q, const float* __restrict__ k,
    const float* __restrict__ ebs, const float* __restrict__ ebt,
    float* __restrict__ scores, float* __restrict__ mbuf, int E)
{
    int e    = blockIdx.x * (blockDim.x >> 5) + (threadIdx.x >> 5);
    int lane = threadIdx.x & 31;
    if (e >= E) return;

    int tgt = ei[e];
    int src = ei[E + e];

    float4 qv = ((const float4*)(q + (size_t)tgt * DMODEL))[lane];
    float4 kv = ((const float4*)(k + (size_t)src * DMODEL))[lane];
    float p = qv.x * kv.x + qv.y * kv.y + qv.z * kv.z + qv.w * kv.w;
    p += __shfl_xor(p, 1, 32);
    p += __shfl_xor(p, 2, 32);           // sum over one head's 16 dims

    int h = lane >> 2;
    float s = p * 0.25f                  // 1/sqrt(DK), DK = 16
            + ebs[(size_t)src * NHEADS + h]
            + ebt[(size_t)tgt * NHEADS + h];

    if ((lane & 3) == 0) {
        scores[(size_t)e * NHEADS + h] = s;
        atomicMax(mbuf + (size_t)tgt * NHEADS + h, s);   // f32 max atomic
    }
}

// ---------------------------------------------------------------------------
// Pass B: ex = exp(score - m[tgt]); store ex in-place; den += ex.
// One thread per (edge, head).
// ---------------------------------------------------------------------------
__global__ void edge_exp_sum(const int* __restrict__ ei, float* __restrict__ scores,
                             const float* __restrict__ mbuf, float* __restrict__ den, int E)
{
    int i = blockIdx.x * blockDim.x + threadIdx.x;
    if (i >= E * NHEADS) return;
    int e = i >> 3, h = i & 7;
    int tgt = ei[e];
    float ex = expf(scores[i] - mbuf[(size_t)tgt * NHEADS + h]);
    scores[i] = ex;
    atomicAdd(den + (size_t)tgt * NHEADS + h, ex);
}

// ---------------------------------------------------------------------------
// Pass C: upd[tgt] += (ex/den) * v[src].  One wave per edge, 4 f32 atomics/lane.
// ---------------------------------------------------------------------------
__global__ __launch_bounds__(256) void edge_scatter(const int* __restrict__ ei,
    const float* __restrict__ scores, const float* __restrict__ den,
    const float* __restrict__ v, float* __restrict__ upd, int E)
{
    int e    = blockIdx.x * (blockDim.x >> 5) + (threadIdx.x >> 5);
    int lane = threadIdx.x & 31;
    if (e >= E) return;

    int tgt = ei[e];
    int src = ei[E + e];
    int h   = lane >> 2;

    float w = scores[(size_t)e * NHEADS + h] / den[(size_t)tgt * NHEADS + h];
    float4 vv = ((const float4*)(v + (size_t)src * DMODEL))[lane];

    float* dst = upd + (size_t)tgt * DMODEL + 4 * lane;
    atomicAdd(dst + 0, w * vv.x);
    atomicAdd(dst + 1, w * vv.y);
    atomicAdd(dst + 2, w * vv.z);
    atomicAdd(dst + 3, w * vv.w);
}

// ---------------------------------------------------------------------------
// out = LayerNorm(x (+ addv)) * g + b.   One wave per row.
// ---------------------------------------------------------------------------
__global__ __launch_bounds__(256) void add_layernorm(const float* __restrict__ x,
    const float* __restrict__ addv, const float* __restrict__ g,
    const float* __restrict__ b, float* __restrict__ out, int n)
{
    int row  = blockIdx.x * (blockDim.x >> 5) + (threadIdx.x >> 5);
    int lane = threadIdx.x & 31;
    if (row >= n) return;

    float4 xv = ((const float4*)(x + (size_t)row * DMODEL))[lane];
    if (addv) {
        float4 av = ((const float4*)(addv + (size_t)row * DMODEL))[lane];
        xv.x += av.x; xv.y += av.y; xv.z += av.z; xv.w += av.w;
    }
    float s  = xv.x + xv.y + xv.z + xv.w;
    float sq = xv.x * xv.x + xv.y * xv.y + xv.z * xv.z + xv.w * xv.w;
    #pragma unroll
    for (int off = 16; off > 0; off >>= 1) {
        s  += __shfl_xor(s,  off, 32);
        sq += __shfl_xor(sq, off, 32);
    }
    float mean = s * (1.f / 128.f);
    float var  = sq * (1.f / 128.f) - mean * mean;
    float inv  = rsqrtf(var + 1e-5f);

    float4 gv = ((const float4*)g)[lane];
    float4 bv = ((const float4*)b)[lane];
    float4 o;
    o.x = (xv.x - mean) * inv * gv.x + bv.x;
    o.y = (xv.y - mean) * inv * gv.y + bv.y;
    o.z = (xv.z - mean) * inv * gv.z + bv.z;
    o.w = (xv.w - mean) * inv * gv.w + bv.w;
    ((float4*)(out + (size_t)row * DMODEL))[lane] = o;
}

// ---------------------------------------------------------------------------
static void run_relation(const float* x_src, const float* x_tgt, const int* ei,
                         const float* Wq, const float* bq,
                         const float* Wk, const float* bk,
                         const float* Wv, const float* bv,
                         const float* We, const float* be,
                         const float* Wo, const float* bo,
                         const float* g, const float* b,
                         float* out,
                         float* q, float* k, float* v, float* scores, float* upd,
                         float* ebs, float* ebt, float* mbuf, float* den,
                         hipStream_t stream)
{
    const dim3 blk(256);
    const int gemm_blocks = N_NODES / 16;
    const int node_waves  = (N_NODES + 7) / 8;
    const int edge_waves  = (N_EDGES + 7) / 8;

    gemm128_wmma<<<gemm_blocks, blk, 0, stream>>>(x_tgt, Wq, bq, q, N_NODES);
    gemm128_wmma<<<gemm_blocks, blk, 0, stream>>>(x_src, Wk, bk, k, N_NODES);
    gemm128_wmma<<<gemm_blocks, blk, 0, stream>>>(x_src, Wv, bv, v, N_NODES);

    edge_bias_proj<<<node_waves, blk, 0, stream>>>(x_src, We, nullptr, ebs, N_NODES);
    edge_bias_proj<<<node_waves, blk, 0, stream>>>(x_tgt, We + DMODEL * NHEADS, be, ebt, N_NODES);

    fill_f32<<<(N_NODES * NHEADS + 255) / 256, blk, 0, stream>>>(mbuf, -__builtin_huge_valf(), N_NODES * NHEADS);
    fill_f32<<<(N_NODES * NHEADS + 255) / 256, blk, 0, stream>>>(den, 0.f, N_NODES * NHEADS);
    fill_f32<<<(N_NODES * DMODEL + 255) / 256, blk, 0, stream>>>(upd, 0.f, N_NODES * DMODEL);

    edge_scores_max<<<edge_waves, blk, 0, stream>>>(ei, q, k, ebs, ebt, scores, mbuf, N_EDGES);
    edge_exp_sum<<<(N_EDGES * NHEADS + 255) / 256, blk, 0, stream>>>(ei, scores, mbuf, den, N_EDGES);
    edge_scatter<<<edge_waves, blk, 0, stream>>>(ei, scores, den, v, upd, N_EDGES);

    // output projection reuses q buffer
    gemm128_wmma<<<gemm_blocks, blk, 0, stream>>>(upd, Wo, bo, q, N_NODES);
    add_layernorm<<<node_waves, blk, 0, stream>>>(x_tgt, q, g, b, out, N_NODES);
}

// ---------------------------------------------------------------------------
extern "C" void kernel_launch(void* const* d_in, const int* in_sizes, int n_in,
                              void* d_out, int out_size, void* d_ws, size_t ws_size,
                              hipStream_t stream)
{
    const float* x_wave  = (const float*)d_in[0];
    const float* x_trans = (const float*)d_in[1];
    const float* x_tgt   = (const float*)d_in[2];
    const int*   ei_wt   = (const int*)d_in[3];
    const int*   ei_tt   = (const int*)d_in[4];
    const float* Wq_trans = (const float*)d_in[5];
    const float* bq_trans = (const float*)d_in[6];
    const float* Wk_wave  = (const float*)d_in[7];
    const float* bk_wave  = (const float*)d_in[8];
    const float* Wv_wave  = (const float*)d_in[9];
    const float* bv_wave  = (const float*)d_in[10];
    const float* Wq_tgt   = (const float*)d_in[11];
    const float* bq_tgt   = (const float*)d_in[12];
    const float* Wk_trans = (const float*)d_in[13];
    const float* bk_trans = (const float*)d_in[14];
    const float* Wv_trans = (const float*)d_in[15];
    const float* bv_trans = (const float*)d_in[16];
    const float* We_wt    = (const float*)d_in[17];
    const float* be_wt    = (const float*)d_in[18];
    const float* We_tt    = (const float*)d_in[19];
    const float* be_tt    = (const float*)d_in[20];
    const float* Wo_trans = (const float*)d_in[21];
    const float* bo_trans = (const float*)d_in[22];
    const float* Wo_tgt   = (const float*)d_in[23];
    const float* bo_tgt   = (const float*)d_in[24];
    const float* g_wave   = (const float*)d_in[25];
    const float* b_wave   = (const float*)d_in[26];
    const float* g_trans  = (const float*)d_in[27];
    const float* b_trans  = (const float*)d_in[28];
    const float* g_tgt    = (const float*)d_in[29];
    const float* b_tgt    = (const float*)d_in[30];

    float* out = (float*)d_out;
    const size_t ND = (size_t)N_NODES * DMODEL;   // 6.4M floats
    const size_t NH = (size_t)N_NODES * NHEADS;   // 400k floats
    const size_t EH = (size_t)N_EDGES * NHEADS;   // 6.4M floats

    float* ws     = (float*)d_ws;
    float* q      = ws;               // ND (also reused for output projection)
    float* k      = q + ND;           // ND
    float* v      = k + ND;           // ND
    float* scores = v + ND;           // EH
    float* upd    = scores + EH;      // ND
    float* ebs    = upd + ND;         // NH
    float* ebt    = ebs + NH;         // NH
    float* mbuf   = ebt + NH;         // NH
    float* den    = mbuf + NH;        // NH   -> total ~134 MB

    // out_wave = LN(x_wave)
    add_layernorm<<<(N_NODES + 7) / 8, dim3(256), 0, stream>>>(
        x_wave, nullptr, g_wave, b_wave, out, N_NODES);

    // relation 1: wave -> transition
    run_relation(x_wave, x_trans, ei_wt,
                 Wq_trans, bq_trans, Wk_wave, bk_wave, Wv_wave, bv_wave,
                 We_wt, be_wt, Wo_trans, bo_trans, g_trans, b_trans,
                 out + ND,
                 q, k, v, scores, upd, ebs, ebt, mbuf, den, stream);

    // relation 2: transition -> target (uses ORIGINAL x_trans as source)
    run_relation(x_trans, x_tgt, ei_tt,
                 Wq_tgt, bq_tgt, Wk_trans, bk_trans, Wv_trans, bv_trans,
                 We_tt, be_tt, Wo_tgt, bo_tgt, g_tgt, b_tgt,
                 out + 2 * ND,
                 q, k, v, scores, upd, ebs, ebt, mbuf, den, stream);
}